// LOBContextSummarizer_45217415692841
// MI455X (gfx1250) — compile-verified
//
#include <hip/hip_runtime.h>
#include <math.h>

typedef __attribute__((ext_vector_type(2))) float v2f;
typedef __attribute__((ext_vector_type(8))) float v8f;

#define B_   64
#define T_   4096
#define L_   10
#define C_   4
#define HID_ 16
#define D_   64
#define HC_  15
#define KP_  80          // padded K for GEMM2 (HC_+D_ = 79 -> 80)
#define EPS_ 1e-9f

// ---------------------------------------------------------------------------
// Kernel 1: per-batch mean of mid price over T  (meanmid[b])
// ---------------------------------------------------------------------------
__global__ __launch_bounds__(256) void midmean_kernel(const float* __restrict__ lob,
                                                      float* __restrict__ meanmid) {
  __shared__ float red[256];
  const int b = blockIdx.x;
  const float* ap = lob + ((size_t)(b * C_ + 0) * T_) * L_;   // ask_p channel
  const float* bp = lob + ((size_t)(b * C_ + 1) * T_) * L_;   // bid_p channel
  float s = 0.f;
  for (int t = threadIdx.x; t < T_; t += 256)
    s += 0.5f * (ap[(size_t)t * L_] + bp[(size_t)t * L_]);
  red[threadIdx.x] = s;
  __syncthreads();
  for (int off = 128; off > 0; off >>= 1) {
    if (threadIdx.x < off) red[threadIdx.x] += red[threadIdx.x + off];
    __syncthreads();
  }
  if (threadIdx.x == 0) meanmid[b] = red[0] * (1.f / (float)T_);
}

// B-fragment swizzle: element (k,n) -> Bsw[(kstep*64 + n)*4 + khalf*2 + klo]
// kstep = k>>2, khalf = (k&3)>>1, klo = k&1.
// A lane with kk = (lane<16 ? 0 : 2) then reads its frag as ONE contiguous
// float2 at ((kstep*64 + ncol)*4 + (kk>>1)*2): a single bank-conflict-free
// ds_load_b64, no register packing movs.
__device__ __forceinline__ int bswz(int k, int n) {
  return (((k >> 2) * 64 + n) << 2) + ((k & 2) << 0) + (k & 1);
}

// ---------------------------------------------------------------------------
// Kernel 2: fused handcraft + attention + GEMM1(LN,GELU) + GEMM2(LN,GELU)
// 64 threads (2 wave32) per block, 64 tokens per block.
// ---------------------------------------------------------------------------
__global__ __launch_bounds__(64) void lob_fused_kernel(
    const float* __restrict__ lob,
    const float* __restrict__ w1, const float* __restrict__ b1,
    const float* __restrict__ w2, const float* __restrict__ b2,
    const float* __restrict__ wf, const float* __restrict__ bf,
    const float* __restrict__ ln1g, const float* __restrict__ ln1b,
    const float* __restrict__ wu, const float* __restrict__ bu,
    const float* __restrict__ ln2g, const float* __restrict__ ln2b,
    const float* __restrict__ meanmid,
    float* __restrict__ out)
{
  // ---- LDS (60 KB total) ----
  __shared__ __align__(16) float comb[64][KP_];  // 20480 B : [hc(15) | learned(64) | pad(1)]
  __shared__ __align__(16) float wuS[KP_ * D_];  // 20480 B : wu frag-swizzled, padded to K=80
  __shared__ __align__(16) float uS[5120];       // 20480 B : flat[64][40] + wf-swizzled[40*64]
  float* flatS = uS;                             //           reused as out-stage[64][64]
  float* wfS   = uS + 2560;
  float* outS  = uS;

  const int tid  = threadIdx.x;     // 0..63, one token per thread
  const int lane = tid & 31;
  const int wave = tid >> 5;

  const int g    = blockIdx.x * 64 + tid;   // global token id
  const int bidx = g >> 12;                 // / T_
  const int t    = g & (T_ - 1);

  // ---- cooperative weight staging into frag-swizzled layout (L2 hits) ----
  for (int i = tid; i < L_ * C_ * D_; i += 64) {
    int k = i >> 6, n = i & 63;
    wfS[bswz(k, n)] = wf[i];
  }
  for (int i = tid; i < KP_ * D_; i += 64) {
    int k = i >> 6, n = i & 63;
    wuS[bswz(k, n)] = (k < HC_ + D_) ? wu[k * D_ + n] : 0.f;
  }

  // ---- per-token LOB load (40 floats into registers) ----
  float ap[L_], bp[L_], av[L_], bv[L_];
  {
    const size_t base = (((size_t)bidx * C_) * T_ + t) * L_;
    const size_t cs   = (size_t)T_ * L_;
#pragma unroll
    for (int l = 0; l < L_; ++l) {
      ap[l] = lob[base + 0 * cs + l];
      bp[l] = lob[base + 1 * cs + l];
      av[l] = lob[base + 2 * cs + l];
      bv[l] = lob[base + 3 * cs + l];
    }
  }

  // ---- handcrafted features ----
  {
    float f[HC_];
    const float mid = 0.5f * (ap[0] + bp[0]);
    const float inv_mid = 1.f / (mid + EPS_);
    f[0] = mid / (meanmid[bidx] + EPS_) - 1.f;
    f[1] = (ap[0] - bp[0]) * inv_mid * 1e4f;
    f[2] = (bv[0] - av[0]) / (bv[0] + av[0] + EPS_);
    float sb3 = 0, sa3 = 0, sb5 = 0, sa5 = 0, tb = 0, ta = 0;
#pragma unroll
    for (int l = 0; l < L_; ++l) {
      tb += bv[l]; ta += av[l];
      if (l < 3) { sb3 += bv[l]; sa3 += av[l]; }
      if (l < 5) { sb5 += bv[l]; sa5 += av[l]; }
    }
    f[3] = (sb3 - sa3) / (sb3 + sa3 + EPS_);
    f[4] = (sb5 - sa5) / (sb5 + sa5 + EPS_);
    f[5] = (tb  - ta ) / (tb  + ta  + EPS_);
    f[6] = log1pf(tb);
    f[7] = log1pf(ta);
    const float inv_ta = 1.f / (ta + EPS_), inv_tb = 1.f / (tb + EPS_);
    float apd = 0, bpd = 0;
#pragma unroll
    for (int l = 0; l < L_; ++l) {
      apd += (ap[l] - ap[0]) * av[l];
      bpd += (bp[0] - bp[l]) * bv[l];
    }
    f[8]  = apd * inv_ta * inv_mid * 1e4f;
    f[9]  = bpd * inv_tb * inv_mid * 1e4f;
    f[10] = bv[0] * inv_tb;
    f[11] = av[0] * inv_ta;
    const float pr = (bv[0] * bp[0] - av[0] * ap[0]) / (bv[0] + av[0] + EPS_);
    f[12] = pr * inv_mid * 1e4f;
    const float db = bv[0] - bv[1], da = av[0] - av[1];
    f[13] = log1pf(fabsf(db)) * (db > 0.f ? 1.f : (db < 0.f ? -1.f : 0.f));
    f[14] = log1pf(fabsf(da)) * (da > 0.f ? 1.f : (da < 0.f ? -1.f : 0.f));
#pragma unroll
    for (int i = 0; i < HC_; ++i) comb[tid][i] = f[i];
  }

  // ---- attention MLP over levels + softmax + weighted flat (register/VALU) ----
  {
    float sc[L_];
#pragma unroll
    for (int l = 0; l < L_; ++l) {
      float s = b2[0];
#pragma unroll
      for (int j = 0; j < HID_; ++j) {
        float h = fmaf(ap[l], w1[0 * HID_ + j],
                  fmaf(bp[l], w1[1 * HID_ + j],
                  fmaf(av[l], w1[2 * HID_ + j],
                  fmaf(bv[l], w1[3 * HID_ + j], b1[j]))));
        s = fmaf(fmaxf(h, 0.f), w2[j], s);
      }
      sc[l] = s;
    }
    float m = sc[0];
#pragma unroll
    for (int l = 1; l < L_; ++l) m = fmaxf(m, sc[l]);
    float den = 0.f;
#pragma unroll
    for (int l = 0; l < L_; ++l) { sc[l] = __expf(sc[l] - m); den += sc[l]; }
    const float invden = 1.f / den;
#pragma unroll
    for (int l = 0; l < L_; ++l) {
      const float w = sc[l] * invden;
      flatS[tid * 40 + l * 4 + 0] = ap[l] * w;
      flatS[tid * 40 + l * 4 + 1] = bp[l] * w;
      flatS[tid * 40 + l * 4 + 2] = av[l] * w;
      flatS[tid * 40 + l * 4 + 3] = bv[l] * w;
    }
  }
  __syncthreads();

  // ---- GEMM1: learned_pre = flat(64x40) @ wf(40x64) via V_WMMA_F32_16X16X4_F32 ----
  const int colLane = lane & 15;
  const int kk      = (lane < 16) ? 0 : 2;      // K split per lane half (ISA 16x4 A layout)
  const int rOff    = (lane < 16) ? 0 : 8;      // C/D row split per lane half
#pragma unroll
  for (int s = 0; s < 2; ++s) {
    const int mbase = wave * 32 + s * 16;
    const int arow  = mbase + colLane;
#pragma unroll
    for (int nt = 0; nt < 4; ++nt) {
      const int ncol = nt * 16 + colLane;
      const int boff = (ncol << 2) + (kk << 0);     // per-lane base into swizzled B
      v8f acc = {};
#pragma unroll
      for (int k0 = 0; k0 < 40; k0 += 4) {
        const v2f a  = *(const v2f*)&flatS[arow * 40 + k0 + kk];            // ds_load_b64
        const v2f bm = *(const v2f*)&wfS[((k0 >> 2) << 8) + boff];          // ds_load_b64
        acc = __builtin_amdgcn_wmma_f32_16x16x4_f32(false, a, false, bm,
                                                    (short)0, acc, false, false);
      }
#pragma unroll
      for (int j = 0; j < 8; ++j)
        comb[mbase + rOff + j][HC_ + ncol] = acc[j];
    }
  }
  __syncthreads();

  // ---- LN1 + GELU (per token, in place in comb[15..78]); zero pad col 79 ----
  {
    float vals[D_];
    float mean = 0.f;
#pragma unroll
    for (int d = 0; d < D_; ++d) { vals[d] = comb[tid][HC_ + d] + bf[d]; mean += vals[d]; }
    mean *= (1.f / D_);
    float var = 0.f;
#pragma unroll
    for (int d = 0; d < D_; ++d) { float c = vals[d] - mean; var = fmaf(c, c, var); }
    var *= (1.f / D_);
    const float inv = rsqrtf(var + 1e-5f);
#pragma unroll
    for (int d = 0; d < D_; ++d) {
      const float y = (vals[d] - mean) * inv * ln1g[d] + ln1b[d];
      comb[tid][HC_ + d] = 0.5f * y * (1.f + erff(y * 0.70710678118f));
    }
    comb[tid][KP_ - 1] = 0.f;
  }
  __syncthreads();

  // ---- GEMM2: out_pre = comb(64x80) @ wu(80x64) via V_WMMA_F32_16X16X4_F32 ----
#pragma unroll
  for (int s = 0; s < 2; ++s) {
    const int mbase = wave * 32 + s * 16;
    const int arow  = mbase + colLane;
#pragma unroll
    for (int nt = 0; nt < 4; ++nt) {
      const int ncol = nt * 16 + colLane;
      const int boff = (ncol << 2) + (kk << 0);
      v8f acc = {};
#pragma unroll
      for (int k0 = 0; k0 < KP_; k0 += 4) {
        const v2f a  = *(const v2f*)&comb[arow][k0 + kk];                   // ds_load_b64
        const v2f bm = *(const v2f*)&wuS[((k0 >> 2) << 8) + boff];          // ds_load_b64
        acc = __builtin_amdgcn_wmma_f32_16x16x4_f32(false, a, false, bm,
                                                    (short)0, acc, false, false);
      }
#pragma unroll
      for (int j = 0; j < 8; ++j)
        outS[(mbase + rOff + j) * 64 + ncol] = acc[j];   // uS reuse: flat/wf are dead
    }
  }
  __syncthreads();

  // ---- LN2 + GELU (per token), stage back to LDS ----
  {
    float vals[D_];
    float mean = 0.f;
#pragma unroll
    for (int d = 0; d < D_; ++d) { vals[d] = outS[tid * 64 + d] + bu[d]; mean += vals[d]; }
    mean *= (1.f / D_);
    float var = 0.f;
#pragma unroll
    for (int d = 0; d < D_; ++d) { float c = vals[d] - mean; var = fmaf(c, c, var); }
    var *= (1.f / D_);
    const float inv = rsqrtf(var + 1e-5f);
#pragma unroll
    for (int d = 0; d < D_; ++d) {
      const float y = (vals[d] - mean) * inv * ln2g[d] + ln2b[d];
      vals[d] = 0.5f * y * (1.f + erff(y * 0.70710678118f));
    }
    __syncthreads();   // all LN2 reads of outS finished before overwrite
#pragma unroll
    for (int d = 0; d < D_; ++d) outS[tid * 64 + d] = vals[d];
  }
  __syncthreads();

  // ---- coalesced block store: 64 tokens x 64 feats are contiguous in out ----
  {
    const size_t outbase = (size_t)blockIdx.x * 64 * D_;
    for (int i = tid; i < 64 * D_; i += 64) out[outbase + i] = outS[i];
  }
}

// ---------------------------------------------------------------------------
extern "C" void kernel_launch(void* const* d_in, const int* in_sizes, int n_in,
                              void* d_out, int out_size, void* d_ws, size_t ws_size,
                              hipStream_t stream) {
  (void)in_sizes; (void)n_in; (void)out_size; (void)ws_size;
  const float* lob  = (const float*)d_in[0];
  const float* w1   = (const float*)d_in[1];
  const float* b1   = (const float*)d_in[2];
  const float* w2   = (const float*)d_in[3];
  const float* b2   = (const float*)d_in[4];
  const float* wf   = (const float*)d_in[5];
  const float* bf   = (const float*)d_in[6];
  const float* ln1g = (const float*)d_in[7];
  const float* ln1b = (const float*)d_in[8];
  const float* wu   = (const float*)d_in[9];
  const float* bu   = (const float*)d_in[10];
  const float* ln2g = (const float*)d_in[11];
  const float* ln2b = (const float*)d_in[12];
  float* out = (float*)d_out;
  float* meanmid = (float*)d_ws;   // 64 floats

  midmean_kernel<<<B_, 256, 0, stream>>>(lob, meanmid);

  const int nblocks = (B_ * T_) / 64;  // 4096
  lob_fused_kernel<<<nblocks, 64, 0, stream>>>(lob, w1, b1, w2, b2, wf, bf,
                                               ln1g, ln1b, wu, bu, ln2g, ln2b,
                                               meanmid, out);
}